// KernelSoftArgmax_89575837925954
// MI455X (gfx1250) — compile-verified
//
#include <hip/hip_runtime.h>
#include <math.h>

// Soft-argmax over a (4,64,64,64,64) correlation volume.
// One 256-thread block per 8 rows; each row = 4096 floats (16 KB).
// Async global->LDS double-buffered streaming (CDNA5 ASYNCcnt path),
// single-pass register softmax, wave32 shuffle reductions.

#define THREADS 256
#define ROWS_PB 8
#define ROW_ELEMS 4096
#define F4_PER_ROW 1024
#define F4_PER_THREAD 4

__device__ __forceinline__ unsigned lds_off32(const void* p) {
  // Generic (flat) pointers to LDS carry the LDS byte offset in addr[31:0]
  // (aperture lives in addr[63:32]) -- see CDNA5 ISA 10.2 aperture mapping.
  return (unsigned)(size_t)p;
}

__device__ __forceinline__ void async_copy_b128(unsigned lds_byte_off,
                                                const float* __restrict__ gbase,
                                                unsigned g_byte_off) {
  // GLOBAL_LOAD_ASYNC_TO_LDS_B128, GVS mode: saddr base + 32-bit vgpr offset.
  // VDST slot carries the per-lane LDS byte address.
  asm volatile("global_load_async_to_lds_b128 %0, %1, %2"
               :: "v"(lds_byte_off), "v"(g_byte_off), "s"(gbase)
               : "memory");
}

__device__ __forceinline__ void combine4(float& m, float& s, float& x, float& y,
                                         float m2, float s2, float x2, float y2) {
  float M = fmaxf(m, m2);
  float a = __builtin_amdgcn_exp2f(m - M);
  float b = __builtin_amdgcn_exp2f(m2 - M);
  s = s * a + s2 * b;
  x = x * a + x2 * b;
  y = y * a + y2 * b;
  m = M;
}

__global__ __launch_bounds__(THREADS)
void KernelSoftArgmax_89575837925954_kernel(const float* __restrict__ corr,
                                            const float* __restrict__ beta_log,
                                            const float* __restrict__ coords,
                                            float* __restrict__ out) {
  __shared__ float4 buf[2][F4_PER_ROW];   // 2 x 16 KB double buffer
  __shared__ float s_coords[64];
  __shared__ float red[4][8];

  const int t = threadIdx.x;
  const float* gbase = corr + (size_t)blockIdx.x * (size_t)(ROWS_PB * ROW_ELEMS);

  // Prologue: kick off async stream of row 0 into buffer 0.
#pragma unroll
  for (int i = 0; i < F4_PER_THREAD; ++i) {
    const unsigned f4 = (unsigned)(i * THREADS + t);
    async_copy_b128(lds_off32(&buf[0][f4]), gbase, f4 * 16u);
  }

  if (t < 64) s_coords[t] = coords[t];
  // p = exp(c/beta) = exp2(c * log2(e)/beta); fold into one scale k.
  const float k = 1.4426950408889634f * __expf(-beta_log[0]);
  __syncthreads();

  // Chunk layout: element index e = i*1024 + t*4  =>
  //   W = (t&15)*4 + j   (row-invariant, per-thread)
  //   H = i*16 + (t>>4)  (row-invariant, per-thread)
  const int wbase = (t & 15) * 4;
  const float cx0 = s_coords[wbase + 0];
  const float cx1 = s_coords[wbase + 1];
  const float cx2 = s_coords[wbase + 2];
  const float cx3 = s_coords[wbase + 3];
  float cyv[F4_PER_THREAD];
#pragma unroll
  for (int i = 0; i < F4_PER_THREAD; ++i) cyv[i] = s_coords[i * 16 + (t >> 4)];

  for (int r = 0; r < ROWS_PB; ++r) {
    const int bsel = r & 1;

    if (r + 1 < ROWS_PB) {
      const int nsel = bsel ^ 1;
#pragma unroll
      for (int i = 0; i < F4_PER_THREAD; ++i) {
        const unsigned f4 = (unsigned)(i * THREADS + t);
        async_copy_b128(lds_off32(&buf[nsel][f4]), gbase,
                        (unsigned)(((r + 1) * ROW_ELEMS + f4 * 4) * 4));
      }
      // 4 newer loads in flight; wait for the 4 oldest (this row's buffer).
      asm volatile("s_wait_asynccnt 0x4" ::: "memory");
    } else {
      asm volatile("s_wait_asynccnt 0x0" ::: "memory");
    }

    // Pass 1: stage 16 scaled values in VGPRs, find thread-local max.
    float tv[16];
    float m = -INFINITY;
#pragma unroll
    for (int i = 0; i < F4_PER_THREAD; ++i) {
      const float4 v = buf[bsel][i * THREADS + t];
      const float t0 = k * v.x, t1 = k * v.y, t2 = k * v.z, t3 = k * v.w;
      tv[4 * i + 0] = t0; tv[4 * i + 1] = t1;
      tv[4 * i + 2] = t2; tv[4 * i + 3] = t3;
      m = fmaxf(m, fmaxf(fmaxf(t0, t1), fmaxf(t2, t3)));
    }

    // Pass 2: exp2 accumulate s, wx, wy (all from registers).
    float s = 0.f, wx = 0.f, wy = 0.f;
#pragma unroll
    for (int i = 0; i < F4_PER_THREAD; ++i) {
      const float p0 = __builtin_amdgcn_exp2f(tv[4 * i + 0] - m);
      const float p1 = __builtin_amdgcn_exp2f(tv[4 * i + 1] - m);
      const float p2 = __builtin_amdgcn_exp2f(tv[4 * i + 2] - m);
      const float p3 = __builtin_amdgcn_exp2f(tv[4 * i + 3] - m);
      const float ps = (p0 + p1) + (p2 + p3);
      s += ps;
      wx = fmaf(p0, cx0, fmaf(p1, cx1, fmaf(p2, cx2, fmaf(p3, cx3, wx))));
      wy = fmaf(cyv[i], ps, wy);
    }

    // Intra-wave online-softmax merge (wave32).
#pragma unroll
    for (int mask = 16; mask >= 1; mask >>= 1) {
      const float m2 = __shfl_xor(m, mask, 32);
      const float s2 = __shfl_xor(s, mask, 32);
      const float x2 = __shfl_xor(wx, mask, 32);
      const float y2 = __shfl_xor(wy, mask, 32);
      combine4(m, s, wx, wy, m2, s2, x2, y2);
    }

    const int wave = t >> 5, lane = t & 31;
    if (lane == 0) {
      red[0][wave] = m; red[1][wave] = s; red[2][wave] = wx; red[3][wave] = wy;
    }
    __syncthreads();

    if (t < 32) {
      const bool valid = (t < 8);
      float mm = valid ? red[0][t] : -INFINITY;
      float ss = valid ? red[1][t] : 0.f;
      float xx = valid ? red[2][t] : 0.f;
      float yy = valid ? red[3][t] : 0.f;
#pragma unroll
      for (int mask = 4; mask >= 1; mask >>= 1) {
        const float m2 = __shfl_xor(mm, mask, 32);
        const float s2 = __shfl_xor(ss, mask, 32);
        const float x2 = __shfl_xor(xx, mask, 32);
        const float y2 = __shfl_xor(yy, mask, 32);
        combine4(mm, ss, xx, yy, m2, s2, x2, y2);
      }
      if (t == 0) {
        const float inv = 1.f / ss;
        const size_t row = (size_t)blockIdx.x * ROWS_PB + (size_t)r;
        out[row * 2 + 0] = xx * inv;
        out[row * 2 + 1] = yy * inv;
      }
    }
    __syncthreads();  // red[] reused next row
  }
}

extern "C" void kernel_launch(void* const* d_in, const int* in_sizes, int n_in,
                              void* d_out, int out_size, void* d_ws, size_t ws_size,
                              hipStream_t stream) {
  (void)n_in; (void)d_ws; (void)ws_size; (void)out_size;
  const float* corr     = (const float*)d_in[0];
  const float* beta_log = (const float*)d_in[1];
  const float* coords   = (const float*)d_in[2];
  float* out = (float*)d_out;

  const int rows = in_sizes[0] / ROW_ELEMS;   // 16384
  const int grid = rows / ROWS_PB;            // 2048 blocks x 256 threads
  KernelSoftArgmax_89575837925954_kernel<<<grid, THREADS, 0, stream>>>(
      corr, beta_log, coords, out);
}